// MMCL_PGD_57372173140631
// MI455X (gfx1250) — compile-verified
//
#include <hip/hip_runtime.h>
#include <hip/hip_bf16.h>
#include <math.h>

typedef __attribute__((ext_vector_type(2))) float v2f;
typedef __attribute__((ext_vector_type(8))) float v8f;

#define BS    256
#define NN    255
#define D     2048
#define GAMMA 0.07f
#define REG   0.1f
#define CCAP  1.0f
#define ITERS 250
#define ETA0  0.001f

// ---------------------------------------------------------------------------
// 1) ftr = z / max(norm_over_views(z), 1e-12); layout (512 x 2048) row-major
// ---------------------------------------------------------------------------
__global__ __launch_bounds__(256) void ftr_kernel(const float* __restrict__ z,
                                                  float* __restrict__ ftr) {
  int idx = blockIdx.x * 256 + threadIdx.x;          // 0 .. 256*2048-1
  int b = idx >> 11, d = idx & 2047;
  float z0 = z[(size_t)b * (2 * D) + d];
  float z1 = z[(size_t)b * (2 * D) + D + d];
  float n = fmaxf(sqrtf(z0 * z0 + z1 * z1), 1e-12f);
  ftr[(size_t)b * D + d]        = z0 / n;
  ftr[(size_t)(b + BS) * D + d] = z1 / n;
}

// ---------------------------------------------------------------------------
// 2) xn[r] = sum_d ftr[r,d]^2  (r = 0..511)
// ---------------------------------------------------------------------------
__global__ __launch_bounds__(256) void rowsq_kernel(const float* __restrict__ ftr,
                                                    float* __restrict__ xn) {
  int r = blockIdx.x;
  float s = 0.f;
  for (int d = threadIdx.x; d < D; d += 256) {
    float f = ftr[(size_t)r * D + d];
    s += f * f;
  }
  __shared__ float red[256];
  red[threadIdx.x] = s; __syncthreads();
  for (int o = 128; o > 0; o >>= 1) {
    if (threadIdx.x < o) red[threadIdx.x] += red[threadIdx.x + o];
    __syncthreads();
  }
  if (threadIdx.x == 0) xn[r] = red[0];
}

// ---------------------------------------------------------------------------
// 3) Gram: K[i,j] = exp(-g*(xn_i + xn_j - 2 dot)),  i<256, j<512.
//    WMMA f32 16x16x4, one 32x32 output tile per wave (2 A frags x 2 B frags,
//    4 WMMA per k-step -> 8 FLOP/byte from L2).
//    Lane L loads float2 at row (base + L%16), col (k + 2*(L/16)).
// ---------------------------------------------------------------------------
__global__ __launch_bounds__(256) void gram_kernel(const float* __restrict__ ftr,
                                                   const float* __restrict__ xn,
                                                   float* __restrict__ K) {
  int wave = threadIdx.x >> 5, lane = threadIdx.x & 31;
  int tile = blockIdx.x * 8 + wave;                  // 0..127
  int ti = tile >> 4, tj = tile & 15;                // 8 x 16 tiles of 32x32
  int i0 = ti * 32, j0 = tj * 32;
  int lr = lane & 15, lh = lane >> 4;
  const float* ap0 = ftr + (size_t)(i0 + lr) * D + 2 * lh;
  const float* ap1 = ftr + (size_t)(i0 + 16 + lr) * D + 2 * lh;
  const float* bp0 = ftr + (size_t)(j0 + lr) * D + 2 * lh;
  const float* bp1 = ftr + (size_t)(j0 + 16 + lr) * D + 2 * lh;
  v8f c00 = {}, c01 = {}, c10 = {}, c11 = {};
#pragma unroll 4
  for (int k = 0; k < D; k += 4) {
    v2f a0 = *(const v2f*)(ap0 + k);
    v2f a1 = *(const v2f*)(ap1 + k);
    v2f b0 = *(const v2f*)(bp0 + k);
    v2f b1 = *(const v2f*)(bp1 + k);
    c00 = __builtin_amdgcn_wmma_f32_16x16x4_f32(false, a0, false, b0, (short)0, c00, false, false);
    c01 = __builtin_amdgcn_wmma_f32_16x16x4_f32(false, a0, false, b1, (short)0, c01, false, false);
    c10 = __builtin_amdgcn_wmma_f32_16x16x4_f32(false, a1, false, b0, (short)0, c10, false, false);
    c11 = __builtin_amdgcn_wmma_f32_16x16x4_f32(false, a1, false, b1, (short)0, c11, false, false);
  }
#pragma unroll
  for (int r = 0; r < 8; ++r) {
    int iA = i0 + r + 8 * lh;       // D-tile: VGPR r -> M=r (lanes0-15), M=r+8
    int iB = iA + 16;
    int jA = j0 + lr;               //          N = lane%16
    int jB = jA + 16;
    K[(size_t)iA * 512 + jA] = __expf(-GAMMA * (xn[iA] + xn[jA] - 2.0f * c00[r]));
    K[(size_t)iA * 512 + jB] = __expf(-GAMMA * (xn[iA] + xn[jB] - 2.0f * c01[r]));
    K[(size_t)iB * 512 + jA] = __expf(-GAMMA * (xn[iB] + xn[jA] - 2.0f * c10[r]));
    K[(size_t)iB * 512 + jB] = __expf(-GAMMA * (xn[iB] + xn[jB] - 2.0f * c11[r]));
  }
}

// ---------------------------------------------------------------------------
// 4) eta[t] = 0.001 / ||Q_t||_F ; Q_t[i,j]=1+0.1d+KK[i,j]-KK[t,j]-KK[t,i], i,j!=t
// ---------------------------------------------------------------------------
__global__ __launch_bounds__(256) void qn_kernel(const float* __restrict__ K,
                                                 float* __restrict__ eta) {
  int t = blockIdx.x, i = threadIdx.x;
  float acc = 0.f;
  if (i != t) {
    float kti = K[(size_t)t * 512 + i];
    const float* rowi = K + (size_t)i * 512;
    const float* rowt = K + (size_t)t * 512;
    for (int j = 0; j < BS; ++j) {
      if (j == t) continue;
      float e = 1.0f + ((i == j) ? REG : 0.0f) + rowi[j] - rowt[j] - kti;
      acc += e * e;
    }
  }
  __shared__ float red[256];
  red[i] = acc; __syncthreads();
  for (int o = 128; o > 0; o >>= 1) {
    if (i < o) red[i] += red[i + o];
    __syncthreads();
  }
  if (i == 0) eta[t] = ETA0 / sqrtf(red[0]);
}

// ---------------------------------------------------------------------------
// 5) init: extended column-major alpha/velocity: X[t*256+a], diag forced to 0
// ---------------------------------------------------------------------------
__global__ __launch_bounds__(256) void init_kernel(const float* __restrict__ alpha0,
                                                   float* __restrict__ A,
                                                   float* __restrict__ V) {
  int t = blockIdx.x, a = threadIdx.x;
  float val = 0.f;
  if (a != t) {
    int i = (a < t) ? a : a - 1;                 // idx[t,i] = a
    val = alpha0[(size_t)t * NN + i];
    val = fminf(fmaxf(val, 0.f), CCAP);
  }
  A[(size_t)t * BS + a] = val;
  V[(size_t)t * BS + a] = val;
}

// ---------------------------------------------------------------------------
// 6) per-iteration GEMM:  M^T = V^T * KK   (KK symmetric, KK = K[:, :256])
//    M stored column-major: M[t*256 + a].  32x32 tile per wave, 64 tiles.
// ---------------------------------------------------------------------------
__global__ __launch_bounds__(256) void pgd_gemm_kernel(const float* __restrict__ V,
                                                       const float* __restrict__ K,
                                                       float* __restrict__ M) {
  int wave = threadIdx.x >> 5, lane = threadIdx.x & 31;
  int tile = blockIdx.x * 8 + wave;                  // 0..63
  int tt = tile >> 3, ta = tile & 7;                 // 8 x 8 tiles of 32x32
  int t0 = tt * 32, a0 = ta * 32;
  int lr = lane & 15, lh = lane >> 4;
  const float* ap0 = V + (size_t)(t0 + lr) * BS + 2 * lh;      // A[m=t][k] = V[k,t]
  const float* ap1 = V + (size_t)(t0 + 16 + lr) * BS + 2 * lh;
  const float* bp0 = K + (size_t)(a0 + lr) * 512 + 2 * lh;     // B[k][n=a] = KK[a][k] (sym)
  const float* bp1 = K + (size_t)(a0 + 16 + lr) * 512 + 2 * lh;
  v8f c00 = {}, c01 = {}, c10 = {}, c11 = {};
#pragma unroll 4
  for (int k = 0; k < BS; k += 4) {
    v2f a0v = *(const v2f*)(ap0 + k);
    v2f a1v = *(const v2f*)(ap1 + k);
    v2f b0v = *(const v2f*)(bp0 + k);
    v2f b1v = *(const v2f*)(bp1 + k);
    c00 = __builtin_amdgcn_wmma_f32_16x16x4_f32(false, a0v, false, b0v, (short)0, c00, false, false);
    c01 = __builtin_amdgcn_wmma_f32_16x16x4_f32(false, a0v, false, b1v, (short)0, c01, false, false);
    c10 = __builtin_amdgcn_wmma_f32_16x16x4_f32(false, a1v, false, b0v, (short)0, c10, false, false);
    c11 = __builtin_amdgcn_wmma_f32_16x16x4_f32(false, a1v, false, b1v, (short)0, c11, false, false);
  }
#pragma unroll
  for (int r = 0; r < 8; ++r) {
    int tA = t0 + r + 8 * lh, tB = tA + 16;
    int aA = a0 + lr,         aB = aA + 16;
    M[(size_t)tA * BS + aA] = c00[r];                // 16-lane contiguous stores
    M[(size_t)tA * BS + aB] = c01[r];
    M[(size_t)tB * BS + aA] = c10[r];
    M[(size_t)tB * BS + aB] = c11[r];
  }
}

// ---------------------------------------------------------------------------
// 7) per-iteration update: g = S*(1-KK[t,a]) + 0.1v + M[a,t] - M[t,t] - 2
//    a_new = clip(v - eta_t*g, 0, 1);  v_next = a_new + beta*(a_new - a_old)
// ---------------------------------------------------------------------------
__global__ __launch_bounds__(256) void pgd_update_kernel(const float* __restrict__ Vin,
                                                         const float* __restrict__ Ain,
                                                         const float* __restrict__ M,
                                                         const float* __restrict__ K,
                                                         const float* __restrict__ eta,
                                                         float beta_next,
                                                         float* __restrict__ Aout,
                                                         float* __restrict__ Vout) {
  int t = blockIdx.x, a = threadIdx.x;
  float v = Vin[(size_t)t * BS + a];                 // diag entries are 0
  __shared__ float red[256];
  red[a] = v; __syncthreads();
  for (int o = 128; o > 0; o >>= 1) {
    if (a < o) red[a] += red[a + o];
    __syncthreads();
  }
  float S   = red[0];
  float m   = M[(size_t)t * BS + a];
  float mtt = M[(size_t)t * BS + t];                 // = sum_a KK[t,a]*v[a,t]
  float kta = K[(size_t)t * 512 + a];
  float g   = S * (1.0f - kta) + REG * v + m - mtt - 2.0f;
  float an  = fminf(fmaxf(v - eta[t] * g, 0.f), CCAP);
  if (a == t) an = 0.f;
  float ao  = Ain[(size_t)t * BS + a];
  float vn  = an + beta_next * (an - ao);
  if (a == t) vn = 0.f;
  Aout[(size_t)t * BS + a] = an;
  Vout[(size_t)t * BS + a] = vn;
}

// ---------------------------------------------------------------------------
// 8) final loss / statistics reductions
//    acc: [0]=neg_sum [1]=kn_sum [2]=posloss_sum [3]=posdiag_sum
//         [4]=#ones [5]=#pos [6]=#zeros
// ---------------------------------------------------------------------------
__global__ __launch_bounds__(32) void zero_acc_kernel(float* __restrict__ acc) {
  if (threadIdx.x < 8) acc[threadIdx.x] = 0.f;
}

__global__ __launch_bounds__(256) void finalize_kernel(const float* __restrict__ A,
                                                       const float* __restrict__ K,
                                                       float* __restrict__ acc) {
  int t = blockIdx.x, a = threadIdx.x;
  float va = A[(size_t)t * BS + a];                  // diag = 0
  float ks = K[(size_t)a * 512 + BS + t];            // Ks[a, t]
  bool off = (a != t);
  float neg  = off ? va * ks : 0.f;
  float kn   = off ? ks : 0.f;
  float ones = (off && va == CCAP) ? 1.f : 0.f;
  float pos  = (off && va > 0.0f)  ? 1.f : 0.f;
  float zer  = (off && va == 0.0f) ? 1.f : 0.f;
  float ax   = va;

  __shared__ float red[256];
  float in[6] = {neg, kn, ones, pos, zer, ax};
  float out[6];
#pragma unroll
  for (int q = 0; q < 6; ++q) {
    red[a] = in[q]; __syncthreads();
    for (int o = 128; o > 0; o >>= 1) {
      if (a < o) red[a] += red[a + o];
      __syncthreads();
    }
    out[q] = red[0]; __syncthreads();
  }
  if (a == 0) {
    float pd = K[(size_t)t * 512 + BS + t];          // pos_diag[t]
    atomicAdd(&acc[0], out[0]);
    atomicAdd(&acc[1], out[1]);
    atomicAdd(&acc[2], out[5] * pd);
    atomicAdd(&acc[3], pd);
    atomicAdd(&acc[4], out[2]);
    atomicAdd(&acc[5], out[3]);
    atomicAdd(&acc[6], out[4]);
  }
}

__global__ __launch_bounds__(32) void writeout_kernel(const float* __restrict__ acc,
                                                      float* __restrict__ out) {
  if (threadIdx.x == 0) {
    float neg_loss = acc[0] / (float)BS;
    float pos_loss = acc[2] / (float)BS;
    out[0] = neg_loss - pos_loss;                    // loss
    out[1] = acc[3] / (float)BS;                     // pos_diag.mean()
    out[2] = acc[1] / (float)(BS * NN);              // Kn.mean()
    out[3] = acc[4] / (acc[5] + 1e-10f);             // sparsity
    out[4] = acc[6] / (float)(BS * NN);              // num_zero
    out[5] = 0.0f;
  }
}

// ---------------------------------------------------------------------------
// host orchestration
// ---------------------------------------------------------------------------
extern "C" void kernel_launch(void* const* d_in, const int* in_sizes, int n_in,
                              void* d_out, int out_size, void* d_ws, size_t ws_size,
                              hipStream_t stream) {
  const float* z      = (const float*)d_in[0];   // (256, 2, 2048)
  const float* alpha0 = (const float*)d_in[1];   // (256, 255, 1)
  float* out = (float*)d_out;                    // 6 scalars

  float* ws = (float*)d_ws;
  const size_t FTR_OFF = 0;                                // 512*2048
  const size_t XN_OFF  = FTR_OFF + (size_t)512 * D;        // 512
  const size_t K_OFF   = XN_OFF  + 512;                    // 256*512
  const size_t ETA_OFF = K_OFF   + (size_t)BS * 512;       // 256
  const size_t A0_OFF  = ETA_OFF + BS;                     // 256*256 each
  const size_t A1_OFF  = A0_OFF  + (size_t)BS * BS;
  const size_t V0_OFF  = A1_OFF  + (size_t)BS * BS;
  const size_t V1_OFF  = V0_OFF  + (size_t)BS * BS;
  const size_t M_OFF   = V1_OFF  + (size_t)BS * BS;
  const size_t ACC_OFF = M_OFF   + (size_t)BS * BS;

  float* ftr = ws + FTR_OFF;
  float* xn  = ws + XN_OFF;
  float* K   = ws + K_OFF;
  float* eta = ws + ETA_OFF;
  float* A[2] = { ws + A0_OFF, ws + A1_OFF };
  float* V[2] = { ws + V0_OFF, ws + V1_OFF };
  float* M   = ws + M_OFF;
  float* acc = ws + ACC_OFF;

  zero_acc_kernel<<<1, 32, 0, stream>>>(acc);
  ftr_kernel<<<(BS * D) / 256, 256, 0, stream>>>(z, ftr);
  rowsq_kernel<<<512, 256, 0, stream>>>(ftr, xn);
  gram_kernel<<<16, 256, 0, stream>>>(ftr, xn, K);      // 128 32x32 tiles
  qn_kernel<<<BS, 256, 0, stream>>>(K, eta);
  init_kernel<<<BS, 256, 0, stream>>>(alpha0, A[0], V[0]);

  for (int it = 0; it < ITERS; ++it) {
    int src = it & 1, dst = src ^ 1;
    float beta_next = (float)(it + 1) / (float)(it + 4);
    pgd_gemm_kernel<<<8, 256, 0, stream>>>(V[src], K, M);   // 64 32x32 tiles
    pgd_update_kernel<<<BS, 256, 0, stream>>>(V[src], A[src], M, K, eta,
                                              beta_next, A[dst], V[dst]);
  }
  // ITERS = 250 (even) -> final alpha in A[0]
  finalize_kernel<<<BS, 256, 0, stream>>>(A[0], K, acc);
  writeout_kernel<<<1, 32, 0, stream>>>(acc, out);
  (void)in_sizes; (void)n_in; (void)out_size; (void)ws_size;
}